// VolumetricSSIM_43310450213283
// MI455X (gfx1250) — compile-verified
//
#include <hip/hip_runtime.h>

// ---------------------------------------------------------------------------
// Volumetric SSIM for MI455X (gfx1250).
//  * 11-tap separable Gaussian blur as banded 16x16x28 fp32 matmuls on the
//    matrix pipes: 7 accumulating v_wmma_f32_16x16x4_f32 per 16-row tile,
//    axes W -> H -> D, for 5 fields (x, y, x^2, y^2, xy).
//  * Guard-free inner loops: K=26,27 band weights are exactly 0, so LDS
//    buffers are padded with finite (zeroed) slack instead of exec-masked
//    loads/stores.
//  * Data staging: TENSOR_LOAD_TO_LDS (TDM) for interior tiles,
//    global_load_async_to_lds_b32 (ASYNCcnt) with zero-fill for border tiles.
//  * SSIM map + deterministic two-stage reduction.
// ---------------------------------------------------------------------------

typedef float v2f __attribute__((ext_vector_type(2)));
typedef float v8f __attribute__((ext_vector_type(8)));
typedef unsigned int v4u __attribute__((ext_vector_type(4)));
typedef int v8i_t __attribute__((ext_vector_type(8)));
typedef int v4i_t __attribute__((ext_vector_type(4)));

#define TILE   16
#define HALO   5
#define HT     26          // TILE + 2*HALO
#define NSTEP  7           // K padded 26 -> 28, 7 steps of K=4
#define NWAVE  8           // 256 threads / wave32

// LDS layout (floats). Pads are zeroed once; all hot-loop accesses unguarded.
//   xt : 26^3 + 8 pad      = 17584
//   yt : 26^3 + 8 pad      = 17584
//   t1 : 688*16 (43 tiles) = 11008   (rows 676..687: finite dump/dup rows)
//   t2 : 28*16*16          =  7168   (dz rows 26,27 zeroed once)
//   rf : 5 * 16^3          = 20480
#define XT_OFF 0
#define YT_OFF 17584
#define T1_OFF 35168
#define T2_OFF 46176
#define RF_OFF 53344
#define SMEM_FLOATS 73824          // 295,296 bytes (< 320KB WGP LDS)

__constant__ float G11[11] = {
    0.00102838f, 0.00759870f, 0.03600077f, 0.10936069f, 0.21300553f,
    0.26601172f,
    0.21300553f, 0.10936069f, 0.03600077f, 0.00759870f, 0.00102838f};

#define C1_SSIM 1.0e-4f
#define C2_SSIM 9.0e-4f

template <int F>
__device__ __forceinline__ float fieldval(const float* __restrict__ xs,
                                          const float* __restrict__ ys,
                                          int idx) {
    if constexpr (F == 0) return xs[idx];
    else if constexpr (F == 1) return ys[idx];
    else if constexpr (F == 2) { float v = xs[idx]; return v * v; }
    else if constexpr (F == 3) { float v = ys[idx]; return v * v; }
    else { return xs[idx] * ys[idx]; }
}

// TDM: one descriptor loads a full 26x26x26 fp32 tile (interior tiles only).
// D# bit layout per CDNA5 ISA ch.8.  data_size=4B, 3D tile, strides 160/25600.
// This toolchain uses the 6-arg builtin: (g0, g1, g2, g3, g4, cpol).
__device__ __forceinline__ void tdm_load_tile(const float* __restrict__ gsrc,
                                              unsigned lds_byte_off) {
    unsigned long long ga = (unsigned long long)(uintptr_t)gsrc;
    const unsigned TD = 1u << 20;   // large tensor extent: no OOB for interior
    v4u g0;
    g0[0] = 1u;                                           // count=1, user desc
    g0[1] = lds_byte_off;                                 // lds_addr
    g0[2] = (unsigned)(ga & 0xFFFFFFFFu);                 // global_addr lo
    g0[3] = (unsigned)((ga >> 32) & 0x1FFFFFFu) | (2u << 30); // hi + type=2
    v8i_t g1;
    g1[0] = (int)(2u << 16);                              // data_size = 4B
    g1[1] = (int)((TD & 0xFFFFu) << 16);                  // tensor_dim0 lo
    g1[2] = (int)(((TD >> 16) & 0xFFFFu) | ((TD & 0xFFFFu) << 16)); // d0hi|d1lo
    g1[3] = (int)(((TD >> 16) & 0xFFFFu) | (26u << 16));  // d1 hi | tile_dim0
    g1[4] = (int)(26u | (26u << 16));                     // tile_dim1|tile_dim2
    g1[5] = 160;                                          // dim0 stride (elems)
    g1[6] = (int)(25600u << 16);                          // s0 hi | s1 lo16
    g1[7] = 0;                                            // s1 hi
    v4i_t g2;
    g2[0] = (int)TD;                                      // tensor_dim2
    g2[1] = 0; g2[2] = 0; g2[3] = 0;                      // dim3 unused
    v4i_t g3; g3[0] = 0; g3[1] = 0; g3[2] = 0; g3[3] = 0;
    v8i_t g4;                                             // extra group (zero)
    g4[0] = 0; g4[1] = 0; g4[2] = 0; g4[3] = 0;
    g4[4] = 0; g4[5] = 0; g4[6] = 0; g4[7] = 0;
    __builtin_amdgcn_tensor_load_to_lds(g0, g1, g2, g3, g4, 0);
}

// One field: blur along W, H, D via v_wmma_f32_16x16x4_f32, no guards.
template <int F>
__device__ __forceinline__ void blur_field(const float* __restrict__ xt,
                                           const float* __restrict__ yt,
                                           float* __restrict__ t1,
                                           float* __restrict__ t2,
                                           float* __restrict__ out,
                                           const v2f* __restrict__ bfrag,
                                           int wave, int lane) {
    const int n     = lane & 15;          // D/B column (N)
    const int khalf = (lane >> 4) << 1;   // A/B K sub-offset: 0 or 2
    const int dhi   = (lane >> 4);        // D row half select

    // ---- Pass A: blur along W. M=(dz,dy): 676 rows -> 43 M-tiles ----------
    for (int mt = wave; mt < 43; mt += NWAVE) {
        int m = mt * 16 + n;
        if (m > 675) m = 675;             // clamp: dup rows land in t1 pad
        int rb = m * 26;                  // (dz*26+dy)*26 == m*26
        v8f c = {};
#pragma unroll
        for (int s = 0; s < NSTEP; ++s) {
            int k0 = 4 * s + khalf;
            v2f a;                        // K=26,27 hit zeroed pad; B there = 0
            a.x = fieldval<F>(xt, yt, rb + k0);
            a.y = fieldval<F>(xt, yt, rb + k0 + 1);
            c = __builtin_amdgcn_wmma_f32_16x16x4_f32(
                    false, a, false, bfrag[s], (short)0, c, false, false);
        }
#pragma unroll
        for (int v = 0; v < 8; ++v) {
            int row = mt * 16 + v + (dhi << 3);   // 0..687, t1 has 688 rows
            t1[row * 16 + n] = c[v];
        }
    }
    __syncthreads();

    // ---- Pass B: blur along H. M=(dz,wx): 416 rows -> 26 M-tiles ----------
    // t1[dz*26+hy][wx] -> t2[dz][oy][wx]; reads at hy=26,27 are finite x 0.
    for (int mt = wave; mt < 26; mt += NWAVE) {
        int m  = mt * 16 + n;
        int dz = m >> 4;
        int wx = m & 15;
        v8f c = {};
#pragma unroll
        for (int s = 0; s < NSTEP; ++s) {
            int k0 = 4 * s + khalf;
            v2f a;
            a.x = t1[(dz * 26 + k0) * 16 + wx];
            a.y = t1[(dz * 26 + k0 + 1) * 16 + wx];
            c = __builtin_amdgcn_wmma_f32_16x16x4_f32(
                    false, a, false, bfrag[s], (short)0, c, false, false);
        }
#pragma unroll
        for (int v = 0; v < 8; ++v) {
            int row = mt * 16 + v + (dhi << 3);   // row = dz*16 + wx
            t2[((row >> 4) * 16 + n) * 16 + (row & 15)] = c[v];
        }
    }
    __syncthreads();

    // ---- Pass C: blur along D. M=(oy,wx): 256 rows -> 16 M-tiles ----------
    // t2[dz][oy][wx] -> out[oz][oy][wx]; dz=26,27 rows zeroed once.
    for (int mt = wave; mt < 16; mt += NWAVE) {
        int m  = mt * 16 + n;
        int oy = m >> 4;
        int wx = m & 15;
        v8f c = {};
#pragma unroll
        for (int s = 0; s < NSTEP; ++s) {
            int k0 = 4 * s + khalf;
            v2f a;
            a.x = t2[(k0 * 16 + oy) * 16 + wx];
            a.y = t2[((k0 + 1) * 16 + oy) * 16 + wx];
            c = __builtin_amdgcn_wmma_f32_16x16x4_f32(
                    false, a, false, bfrag[s], (short)0, c, false, false);
        }
#pragma unroll
        for (int v = 0; v < 8; ++v) {
            int row = mt * 16 + v + (dhi << 3);   // row = oy*16 + wx
            out[n * 256 + row] = c[v];            // n is oz
        }
    }
    __syncthreads();
}

__global__ void __launch_bounds__(256)
ssim3d_wmma_kernel(const float* __restrict__ x, const float* __restrict__ y,
                   float* __restrict__ partial) {
    extern __shared__ float smem[];
    float* xt = smem + XT_OFF;
    float* yt = smem + YT_OFF;
    float* t1 = smem + T1_OFF;
    float* t2 = smem + T2_OFF;
    float* rf = smem + RF_OFF;

    const int tid  = threadIdx.x;
    const int lane = tid & 31;
    const int wave = tid >> 5;

    // 2000 blocks: batch (2) x 10x10x10 tiles
    int b     = blockIdx.x;
    int batch = b / 1000;
    int rm    = b - batch * 1000;
    int tz    = rm / 100;
    int tyy   = (rm / 10) % 10;
    int tx    = rm % 10;
    const long base = (long)batch * 4096000L;   // 160^3
    const int z0 = tz * TILE - HALO;
    const int y0 = tyy * TILE - HALO;
    const int x0 = tx * TILE - HALO;
    const bool interior = (tz >= 1) & (tz <= 8) & (tyy >= 1) & (tyy <= 8) &
                          (tx >= 1) & (tx <= 8);

    // --- Banded Gaussian B fragments (28x16), kept in VGPRs all kernel -----
    v2f bfrag[NSTEP];
    {
        const int n     = lane & 15;
        const int khalf = (lane >> 4) << 1;
#pragma unroll
        for (int s = 0; s < NSTEP; ++s) {
            int k0 = 4 * s + khalf;
            int d0 = k0 - n;
            int d1 = k0 + 1 - n;
            bfrag[s].x = (d0 >= 0 && d0 < 11) ? G11[d0] : 0.0f;
            bfrag[s].y = (d1 >= 0 && d1 < 11) ? G11[d1] : 0.0f;
        }
    }

    // --- Zero phase: small pads always; full x/y tiles only at borders -----
    if (tid < 8) {
        xt[17576 + tid] = 0.0f;
        yt[17576 + tid] = 0.0f;
    }
    for (int i = tid; i < 512; i += 256)        // t2 dz rows 26,27
        t2[28 * 16 * 16 - 512 + i] = 0.0f;
    if (!interior) {
        float4* p = (float4*)smem;              // xt+yt contiguous, 16B aligned
        float4 z4 = make_float4(0.f, 0.f, 0.f, 0.f);
        for (int i = tid; i < (2 * 17584) / 4; i += 256) p[i] = z4;
    }
    __syncthreads();

    // --- Stage haloed x/y tiles into LDS -----------------------------------
    if (interior) {
        // Tensor Data Mover: one 26x26x26 DMA per input, issued by wave 0
        // (TDM ignores EXEC -> exactly one wave must issue).
        if (wave == 0) {
            const float* gx0 = x + base + (((long)z0 * 160 + y0) * 160 + x0);
            const float* gy0 = y + base + (((long)z0 * 160 + y0) * 160 + x0);
            tdm_load_tile(gx0, (unsigned)(uintptr_t)xt);
            tdm_load_tile(gy0, (unsigned)(uintptr_t)yt);
            __builtin_amdgcn_s_wait_tensorcnt(0);
        }
    } else {
        // Border: EXEC-masked async loads; OOB voxels keep their zeros.
        for (int i = tid; i < HT * HT * HT; i += 256) {
            int dz = i / (HT * HT);
            int r2 = i - dz * (HT * HT);
            int dy = r2 / HT;
            int dx = r2 - dy * HT;
            int gz = z0 + dz, gy = y0 + dy, gx = x0 + dx;
            bool ok = (unsigned)gz < 160u && (unsigned)gy < 160u &&
                      (unsigned)gx < 160u;
            if (ok) {
                long gi = base + ((long)gz * 160 + gy) * 160 + gx;
                unsigned lx = (unsigned)(uintptr_t)(xt + i);
                unsigned ly = (unsigned)(uintptr_t)(yt + i);
                asm volatile("global_load_async_to_lds_b32 %0, %1, off"
                             :: "v"(lx), "v"(x + gi) : "memory");
                asm volatile("global_load_async_to_lds_b32 %0, %1, off"
                             :: "v"(ly), "v"(y + gi) : "memory");
            }
        }
        asm volatile("s_wait_asynccnt 0x0" ::: "memory");
    }
    __syncthreads();

    // --- Five blurred fields through the matrix pipes ----------------------
    blur_field<0>(xt, yt, t1, t2, rf + 0 * 4096, bfrag, wave, lane); // mu_x
    blur_field<1>(xt, yt, t1, t2, rf + 1 * 4096, bfrag, wave, lane); // mu_y
    blur_field<2>(xt, yt, t1, t2, rf + 2 * 4096, bfrag, wave, lane); // E[x^2]
    blur_field<3>(xt, yt, t1, t2, rf + 3 * 4096, bfrag, wave, lane); // E[y^2]
    blur_field<4>(xt, yt, t1, t2, rf + 4 * 4096, bfrag, wave, lane); // E[xy]

    // --- SSIM map + block-local reduction ----------------------------------
    float sum = 0.0f;
    for (int i = tid; i < 4096; i += 256) {
        float mx = rf[i];
        float my = rf[4096 + i];
        float xx = rf[2 * 4096 + i];
        float yy = rf[3 * 4096 + i];
        float xy = rf[4 * 4096 + i];
        float mxs = mx * mx, mys = my * my, mxy = mx * my;
        float sx = xx - mxs, sy = yy - mys, sxy = xy - mxy;
        float num = (2.0f * mxy + C1_SSIM) * (2.0f * sxy + C2_SSIM);
        float den = (mxs + mys + C1_SSIM) * (sx + sy + C2_SSIM);
        sum += num / den;
    }
    __syncthreads();              // done reading rf; reuse t1 for reduction
    t1[tid] = sum;
    __syncthreads();
#pragma unroll
    for (int off = 128; off > 0; off >>= 1) {
        if (tid < off) t1[tid] += t1[tid + off];
        __syncthreads();
    }
    if (tid == 0) partial[blockIdx.x] = t1[0];
}

// Deterministic fixed-order final reduction (single block).
__global__ void __launch_bounds__(256)
ssim3d_reduce_kernel(const float* __restrict__ partial, float* __restrict__ out,
                     int nblk, float scale) {
    __shared__ float s[256];
    int tid = threadIdx.x;
    float acc = 0.0f;
    for (int i = tid; i < nblk; i += 256) acc += partial[i];
    s[tid] = acc;
    __syncthreads();
#pragma unroll
    for (int off = 128; off > 0; off >>= 1) {
        if (tid < off) s[tid] += s[tid + off];
        __syncthreads();
    }
    if (tid == 0) out[0] = s[0] * scale;
}

extern "C" void kernel_launch(void* const* d_in, const int* in_sizes, int n_in,
                              void* d_out, int out_size, void* d_ws, size_t ws_size,
                              hipStream_t stream) {
    const float* x = (const float*)d_in[0];
    const float* y = (const float*)d_in[1];
    float* partial = (float*)d_ws;          // needs 2000 * 4 = 8000 bytes
    float* out     = (float*)d_out;

    constexpr int    NBLK  = 2000;          // 2 batches * 10^3 tiles
    constexpr size_t SHMEM = (size_t)SMEM_FLOATS * sizeof(float); // 295,296 B

    ssim3d_wmma_kernel<<<NBLK, 256, SHMEM, stream>>>(x, y, partial);
    ssim3d_reduce_kernel<<<1, 256, 0, stream>>>(partial, out, NBLK,
                                                1.0f / 8192000.0f);
}